// NVSM_25735444037692
// MI455X (gfx1250) — compile-verified
//
#include <hip/hip_runtime.h>
#include <hip/hip_bf16.h>

typedef float v2f __attribute__((ext_vector_type(2)));
typedef float v8f __attribute__((ext_vector_type(8)));

#define LAMB 0.01f

// ---------------------------------------------------------------- utilities
__device__ __forceinline__ float block_reduce(float v, float* sred) {
  const int tid = threadIdx.x;
  sred[tid] = v;
  __syncthreads();
  for (int s = blockDim.x >> 1; s > 0; s >>= 1) {
    if (tid < s) sred[tid] += sred[tid + s];
    __syncthreads();
  }
  float r = sred[0];
  __syncthreads();
  return r;
}

// ---------------------------------------------------------------- kernel 0
__global__ void k_init(float* out) {
  if (threadIdx.x == 0 && blockIdx.x == 0) out[0] = 0.0f;
}

// ---------------------------------------------------------------- kernel 1
// g = rv[:, word_ids].mean(axis=2); normed = g / ||g||_col
// normed stored K x N row-major (WE x B) so GEMM B-loads coalesce.
__global__ void k_gather_norm(const float* __restrict__ rv,
                              const int* __restrict__ word_ids,
                              float* __restrict__ normed,
                              int V, int WE, int B, int NG) {
  const int b   = blockIdx.x;     // batch column
  const int tid = threadIdx.x;    // 128 threads
  __shared__ int   wid[32];
  __shared__ float sred[128];
  if (tid < NG) wid[tid] = word_ids[b * NG + tid];
  __syncthreads();

  const float inv_ng = 1.0f / (float)NG;
  float g[4];
  float ss = 0.0f;
  #pragma unroll
  for (int i = 0; i < 4; ++i) {
    g[i] = 0.0f;
    const int r = tid + i * 128;
    if (r < WE) {
      float s = 0.0f;
      for (int k = 0; k < NG; ++k) s += rv[(size_t)r * V + wid[k]];
      g[i] = s * inv_ng;
      ss += g[i] * g[i];
    }
  }
  sred[tid] = ss;
  __syncthreads();
  for (int s = 64; s > 0; s >>= 1) {
    if (tid < s) sred[tid] += sred[tid + s];
    __syncthreads();
  }
  const float inv_norm = 1.0f / sqrtf(sred[0]);
  #pragma unroll
  for (int i = 0; i < 4; ++i) {
    const int r = tid + i * 128;
    if (r < WE) normed[(size_t)r * B + b] = g[i] * inv_norm;
  }
}

// ---------------------------------------------------------------- kernel 2
// tpre(MxN) = proj(MxK) @ normed(KxN) via V_WMMA_F32_16X16X4_F32.
// One 16x16 tile per wave, 8 waves per 256-thread block.
__global__ void k_gemm_wmma(const float* __restrict__ A,   // M x K  (proj)
                            const float* __restrict__ Bm,  // K x N  (normed)
                            float* __restrict__ C,         // M x N  (tpre)
                            int M, int N, int K) {
  const int lane = threadIdx.x & 31;
  const int wave = threadIdx.x >> 5;
  const int NT = N >> 4;
  const int MT = M >> 4;
  const int tile = blockIdx.x * (blockDim.x >> 5) + wave;
  if (tile >= MT * NT) return;            // wave-uniform: EXEC stays all-ones
  const int m0   = (tile / NT) << 4;
  const int n0   = (tile % NT) << 4;
  const int half = lane >> 4;             // 0: lanes 0-15, 1: lanes 16-31
  const int l16  = lane & 15;
  const int row  = m0 + l16;              // A-matrix row for this lane

  v8f acc = {};
  for (int k0 = 0; k0 + 3 < K; k0 += 4) {
    // A 16x4 fragment: VGPR0 holds K=k0 (lanes 0-15) / K=k0+2 (lanes 16-31),
    //                  VGPR1 holds K=k0+1 / K=k0+3.
    const int ka = k0 + (half << 1);
    v2f a, b;
    a.x = A[(size_t)row * K + ka];
    a.y = A[(size_t)row * K + ka + 1];
    // B 4x16 fragment: rows striped across lanes within a VGPR (K split at 2).
    b.x = Bm[(size_t)ka * N + n0 + l16];
    b.y = Bm[(size_t)(ka + 1) * N + n0 + l16];
    acc = __builtin_amdgcn_wmma_f32_16x16x4_f32(
        /*neg_a=*/false, a, /*neg_b=*/false, b,
        /*c_mod=*/(short)0, acc, /*reuse_a=*/false, /*reuse_b=*/false);
  }
  // C layout: VGPR v -> M = m0 + v + 8*half, N = n0 + l16
  #pragma unroll
  for (int v = 0; v < 8; ++v) {
    C[(size_t)(m0 + v + (half << 3)) * N + n0 + l16] = acc[v];
  }
}

// ---------------------------------------------------------------- kernel 3
// Per-row (axis=1) mean and 1/sqrt(std) with ddof=1 (reference divides by
// sqrt(std), i.e. var^(1/4) -- replicated faithfully).
__global__ void k_row_stats(const float* __restrict__ tpre,
                            float* __restrict__ meanv,
                            float* __restrict__ rfac, int N) {
  const int r   = blockIdx.x;
  const int tid = threadIdx.x;   // 256
  float s = 0.0f, s2 = 0.0f;
  for (int j = tid; j < N; j += blockDim.x) {
    const float v = tpre[(size_t)r * N + j];
    s += v;
    s2 += v * v;
  }
  __shared__ float sa[256], sb[256];
  sa[tid] = s; sb[tid] = s2;
  __syncthreads();
  for (int k = 128; k > 0; k >>= 1) {
    if (tid < k) { sa[tid] += sa[tid + k]; sb[tid] += sb[tid + k]; }
    __syncthreads();
  }
  if (tid == 0) {
    const float m   = sa[0] / (float)N;
    const float var = (sb[0] - (float)N * m * m) / (float)(N - 1);
    const float stdv = sqrtf(fmaxf(var, 0.0f));
    meanv[r] = m;
    rfac[r]  = 1.0f / sqrtf(stdv);
  }
}

// ---------------------------------------------------------------- kernel 4
// Per batch element: t = clip((tpre - mean)/sqrt(std) + beta, -1, 1),
// positive + Z negative sigmoid/log terms, scaled atomic accumulate.
__global__ void k_score(const float* __restrict__ tpre,
                        const float* __restrict__ meanv,
                        const float* __restrict__ rfac,
                        const float* __restrict__ beta,
                        const float* __restrict__ rd,
                        const int* __restrict__ doc_ids,
                        const int* __restrict__ neg_ids,
                        float* __restrict__ out,
                        int D, int B, int Z) {
  const int b = blockIdx.x;
  const int r = threadIdx.x;     // DE = 256 threads
  __shared__ float sred[256];

  float t = tpre[(size_t)r * B + b];
  t = (t - meanv[r]) * rfac[r] + beta[r];
  t = fminf(1.0f, fmaxf(-1.0f, t));

  // positive document
  const int did = doc_ids[b];
  const float dpos = block_reduce(t * rd[(size_t)r * D + did], sred);
  float total = 0.0f;
  if (r == 0) {
    float p = 1.0f / (1.0f + expf(-dpos));
    p = fminf(0.999f, fmaxf(-0.999f, p));
    total = (float)Z * logf(p);
  }
  // negatives
  for (int z = 0; z < Z; ++z) {
    const int nid = neg_ids[b * Z + z];
    const float dneg = block_reduce(t * rd[(size_t)r * D + nid], sred);
    if (r == 0) {
      float p = 1.0f / (1.0f + expf(-dneg));
      p = fminf(0.999f, fmaxf(-0.999f, p));
      total += logf(fmaxf(1.0f - p, 0.01f));
    }
  }
  if (r == 0) {
    const float w = (float)(Z + 1) / (2.0f * (float)Z);
    atomicAdd(out, total * w / (float)B);
  }
}

// ---------------------------------------------------------------- kernel 5
// scale * (sum(rd^2) + sum(proj^2)) -- the dominant 102 MB stream; float4.
__global__ void k_reg(const float* __restrict__ rd, long n_rd,
                      const float* __restrict__ proj, long n_proj,
                      float* __restrict__ out, float scale) {
  const long gid    = blockIdx.x * (long)blockDim.x + threadIdx.x;
  const long stride = gridDim.x * (long)blockDim.x;
  float s = 0.0f;
  const float4* rd4 = (const float4*)rd;
  const long n4 = n_rd >> 2;
  for (long i = gid; i < n4; i += stride) {
    const float4 v = rd4[i];
    s += v.x * v.x + v.y * v.y + v.z * v.z + v.w * v.w;
  }
  for (long i = (n4 << 2) + gid; i < n_rd; i += stride) {
    const float v = rd[i]; s += v * v;
  }
  for (long i = gid; i < n_proj; i += stride) {
    const float v = proj[i]; s += v * v;
  }
  __shared__ float sred[256];
  const float tot = block_reduce(s, sred);
  if (threadIdx.x == 0) atomicAdd(out, tot * scale);
}

// ---------------------------------------------------------------- launcher
extern "C" void kernel_launch(void* const* d_in, const int* in_sizes, int n_in,
                              void* d_out, int out_size, void* d_ws, size_t ws_size,
                              hipStream_t stream) {
  const float* rv       = (const float*)d_in[0];  // (WE, V)
  const float* rd       = (const float*)d_in[1];  // (DE, D)
  const float* proj     = (const float*)d_in[2];  // (DE, WE)
  const float* beta     = (const float*)d_in[3];  // (DE,)
  const int*   word_ids = (const int*)d_in[4];    // (B, NG)
  const int*   doc_ids  = (const int*)d_in[5];    // (B,)
  const int*   neg_ids  = (const int*)d_in[6];    // (B, Z)

  const int DE = in_sizes[3];
  const int WE = in_sizes[2] / DE;
  const int V  = in_sizes[0] / WE;
  const int D  = in_sizes[1] / DE;
  const int B  = in_sizes[5];
  const int NG = in_sizes[4] / B;
  const int Z  = in_sizes[6] / B;

  float* out = (float*)d_out;
  float* ws  = (float*)d_ws;
  float* normed = ws;                         // WE * B
  float* tpre   = normed + (size_t)WE * B;    // DE * B
  float* meanv  = tpre + (size_t)DE * B;      // DE
  float* rfac   = meanv + DE;                 // DE

  k_init<<<1, 1, 0, stream>>>(out);
  k_gather_norm<<<B, 128, 0, stream>>>(rv, word_ids, normed, V, WE, B, NG);

  const int tiles = (DE / 16) * (B / 16);
  k_gemm_wmma<<<(tiles + 7) / 8, 256, 0, stream>>>(proj, normed, tpre, DE, B, WE);

  k_row_stats<<<DE, 256, 0, stream>>>(tpre, meanv, rfac, B);
  k_score<<<B, DE, 0, stream>>>(tpre, meanv, rfac, beta, rd, doc_ids, neg_ids,
                                out, D, B, Z);
  k_reg<<<2048, 256, 0, stream>>>(rd, (long)in_sizes[1], proj, (long)in_sizes[2],
                                  out, LAMB / (2.0f * (float)B));
}